// HEALPixPadding_46162308497902
// MI455X (gfx1250) — compile-verified
//
#include <hip/hip_runtime.h>
#include <cstdint>

// ---------------------------------------------------------------------------
// HEALPix padding (p=2) for data[96, 256, 64, 64] -> out[96, 256, 68, 68].
// Pure data movement: roofline = (400MB in + 454MB out)/23.3TB/s ~ 37us.
// Interior (64x64 of each 68x68 tile) = 88.5% of output -> vectorized b64
// copy, staged through LDS with gfx1250 async instructions
// (global_load_async_to_lds_b64 / global_store_async_from_lds_b64, tracked
// by ASYNCcnt). Halo (528 cells/tile) = scalar gather.
// ---------------------------------------------------------------------------

namespace {
constexpr int N    = 64;            // H == W
constexpr int P    = 2;             // pad width
constexpr int HP   = N + 2 * P;     // 68
constexpr int C    = 256;
constexpr int NF   = 96;            // batch*12 faces
constexpr int NFC  = NF * C;        // 24576 images
constexpr int HALO = HP * HP - N * N; // 528 halo cells per image
}

#define AS1 __attribute__((address_space(1)))
#define AS3 __attribute__((address_space(3)))

typedef int v2i __attribute__((ext_vector_type(2)));  // b64 payload type

// ---------------------------------------------------------------------------
// Interior copy: one block per (face,channel) image. 256 threads x 8 float2
// transfers = 4096 floats = the whole 64x64 tile. 8 independent b64 async
// loads per thread are issued before the single ASYNCcnt wait so each wave
// keeps 2KB in flight (needed to cover HBM latency at 23.3 TB/s). Output row
// starts are 8B-aligned (offset 2 floats, stride 68 floats) so b64 is the
// widest legal vector width.
// ---------------------------------------------------------------------------
__global__ void hpx_interior_kernel(const float* __restrict__ in,
                                    float* __restrict__ out) {
  const int t = threadIdx.x;                 // 0..255
  const size_t nfc = blockIdx.x;             // image index
  const float* gsrc0 = in + nfc * (size_t)(N * N);
  float* gdst0 = out + nfc * (size_t)(HP * HP);

#if defined(__HIP_DEVICE_COMPILE__) && defined(__gfx1250__) &&                 \
    __has_builtin(__builtin_amdgcn_global_load_async_to_lds_b64) &&            \
    __has_builtin(__builtin_amdgcn_global_store_async_from_lds_b64) &&         \
    __has_builtin(__builtin_amdgcn_s_wait_asynccnt)
  // gfx1250 async path: global -> LDS -> global, tracked by ASYNCcnt.
  __shared__ float buf[N * N];               // 16 KB staging per block
#pragma unroll
  for (int k = 0; k < 8; ++k) {
    const int j = k * 256 + t;               // float2 index in [0, 2048)
    __builtin_amdgcn_global_load_async_to_lds_b64(
        (AS1 v2i*)(uintptr_t)(gsrc0 + 2 * j),
        (AS3 v2i*)(uint32_t)(uintptr_t)(&buf[2 * j]), 0, 0);
  }
  __builtin_amdgcn_s_wait_asynccnt(0);       // this wave's LDS fills done
#pragma unroll
  for (int k = 0; k < 8; ++k) {
    const int j = k * 256 + t;
    const int r = j >> 5;                    // row 0..63
    const int v = j & 31;                    // float2 column 0..31
    __builtin_amdgcn_global_store_async_from_lds_b64(
        (AS1 v2i*)(uintptr_t)(gdst0 + (size_t)(r + P) * HP + P + 2 * v),
        (AS3 v2i*)(uint32_t)(uintptr_t)(&buf[2 * j]), 0, 0);
  }
  // S_ENDPGM performs an implicit wait-idle: async stores retire before exit.
#else
  // Fallback: plain VGPR copy, 8 b64 loads in flight per thread.
  float2 vals[8];
#pragma unroll
  for (int k = 0; k < 8; ++k) {
    const int j = k * 256 + t;
    vals[k] = reinterpret_cast<const float2*>(gsrc0)[j];
  }
#pragma unroll
  for (int k = 0; k < 8; ++k) {
    const int j = k * 256 + t;
    const int r = j >> 5;
    const int v = j & 31;
    *reinterpret_cast<float2*>(gdst0 + (size_t)(r + P) * HP + P + 2 * v) =
        vals[k];
  }
#endif
}

// ---------------------------------------------------------------------------
// Halo gather. For output cell (face, y, x) in the 68x68 tile, produce the
// source face + flat N*N index (and optional second source for the blended
// diagonals of the synthesized equatorial corners).
// Conventions (verified against jnp.rot90 with axes=(-2,-1)):
//   rot90(m, 1)[i,j] = m[j, N-1-i]
//   rot90(m, 2)[i,j] = m[N-1-i, N-1-j]
//   rot90(m,-1)[i,j] = m[N-1-j, i]
// ---------------------------------------------------------------------------
__device__ __forceinline__ void hpx_src(int face, int y, int x, int& fA,
                                        int& iA, int& fB, int& iB) {
  fB = -1;
  iB = 0;
  const int ry = (y < P) ? 0 : (y < P + N ? 1 : 2);
  const int rx = (x < P) ? 0 : (x < P + N ? 1 : 2);
  const int yi = y - P, xi = x - P;          // middle-band coords
  const int yb = y - P - N, xr = x - P - N;  // bottom/right band coords

  if (face < 4) {  // ----- north polar faces, _pn -----
    const int i = face;
    const int t = (i + 1) & 3;
    const int tl = (i + 2) & 3;
    const int lf = (i + 3) & 3;
    const int bl = lf;
    const int bf = i + 4;
    const int br = i + 8;
    const int rg = 4 + ((i + 1) & 3);
    const int tr = (i + 1) & 3;
    if (ry == 1 && rx == 1)      { fA = i;  iA = yi * N + xi; }
    else if (ry == 0 && rx == 1) { fA = t;  iA = xi * N + (P - 1 - y); }      // rot90(t,1)
    else if (ry == 2 && rx == 1) { fA = bf; iA = yb * N + xi; }
    else if (ry == 1 && rx == 0) { fA = lf; iA = (P - 1 - x) * N + yi; }      // rot90(lft,-1)
    else if (ry == 1 && rx == 2) { fA = rg; iA = yi * N + xr; }
    else if (ry == 0 && rx == 0) { fA = tl; iA = (P - 1 - y) * N + (P - 1 - x); } // rot90(tl,2)
    else if (ry == 2 && rx == 0) { fA = bl; iA = yb * N + (N - P + x); }
    else if (ry == 0 && rx == 2) { fA = tr; iA = (N - P + y) * N + xr; }
    else                         { fA = br; iA = yb * N + xr; }
  } else if (face < 8) {  // ----- equatorial faces, _pe -----
    const int k = face - 4;
    const int t = k;
    const int lf = (k + 3) & 3;
    const int bl = 4 + ((k + 3) & 3);
    const int bf = 8 + ((k + 3) & 3);
    const int rg = 8 + k;
    const int tr = 4 + ((k + 1) & 3);
    if (ry == 1 && rx == 1)      { fA = face; iA = yi * N + xi; }
    else if (ry == 0 && rx == 1) { fA = t;  iA = (N - P + y) * N + xi; }
    else if (ry == 2 && rx == 1) { fA = bf; iA = yb * N + xi; }
    else if (ry == 1 && rx == 0) { fA = lf; iA = yi * N + (N - P + x); }
    else if (ry == 1 && rx == 2) { fA = rg; iA = yi * N + xr; }
    else if (ry == 2 && rx == 0) { fA = bl; iA = yb * N + (N - P + x); }
    else if (ry == 0 && rx == 2) { fA = tr; iA = (N - P + y) * N + xr; }
    else if (ry == 0 && rx == 0) {
      // synthesized _tl(top=t, lft=lf) corner, y,x in [0,P)
      if (y < x)      { fA = t;  iA = (N - P + y) * N + (x - 1 - y); }
      else if (y > x) { fA = lf; iA = (y - 1 - x) * N + (N - P + x); }
      else {  // blended diagonal: 0.5*top[N-P+y,0] + 0.5*lft[0,N-P+y]
        fA = t;  iA = (N - P + y) * N + 0;
        fB = lf; iB = 0 * N + (N - P + y);
      }
    } else {
      // synthesized _br(b=bf, r=rg) corner, r=yb, c=xr in [0,P)
      const int r = yb, c2 = xr;
      if (r > c2)      { fA = bf; iA = r * N + (N - r + c2); }
      else if (r < c2) { fA = rg; iA = (N - c2 + r) * N + c2; }
      else {  // blended diagonal: 0.5*b[r,N-1] + 0.5*r[N-1,r]
        fA = bf; iA = r * N + (N - 1);
        fB = rg; iB = (N - 1) * N + r;
      }
    }
  } else {  // ----- south polar faces, _ps -----
    const int k = face - 8;
    const int t = 4 + ((k + 1) & 3);
    const int tl = k;
    const int lf = 4 + k;
    const int bl = 8 + ((k + 3) & 3);
    const int bf = bl;
    const int br = 8 + ((k + 2) & 3);
    const int rg = 8 + ((k + 1) & 3);
    const int tr = rg;
    if (ry == 1 && rx == 1)      { fA = face; iA = yi * N + xi; }
    else if (ry == 0 && rx == 1) { fA = t;  iA = (N - P + y) * N + xi; }
    else if (ry == 2 && rx == 1) { fA = bf; iA = xi * N + (N - 1 - yb); }     // rot90(b,1)
    else if (ry == 1 && rx == 0) { fA = lf; iA = yi * N + (N - P + x); }
    else if (ry == 1 && rx == 2) { fA = rg; iA = (N - 1 - xr) * N + yi; }     // rot90(rgt,-1)
    else if (ry == 0 && rx == 0) { fA = tl; iA = (N - P + y) * N + (N - P + x); }
    else if (ry == 2 && rx == 0) { fA = bl; iA = yb * N + (N - P + x); }
    else if (ry == 0 && rx == 2) { fA = tr; iA = (N - P + y) * N + xr; }
    else                         { fA = br; iA = (N - 1 - yb) * N + (N - 1 - xr); } // rot90(br,2)
  }
}

__global__ void hpx_halo_kernel(const float* __restrict__ in,
                                float* __restrict__ out, int total) {
  const int tid = blockIdx.x * blockDim.x + threadIdx.x;
  if (tid >= total) return;
  const int h = tid % HALO;    // halo cell within one image
  const int nfc = tid / HALO;  // (b*12 + face)*C + c
  const int face = (nfc / C) % 12;

  // decode h -> (y, x): top band, bottom band, left band, right band
  int y, x;
  if (h < 2 * HP) {              // 136: rows 0..1, all x
    y = h / HP;
    x = h % HP;
  } else if (h < 4 * HP) {       // 136: rows 66..67, all x
    const int u = h - 2 * HP;
    y = (HP - P) + u / HP;
    x = u % HP;
  } else if (h < 4 * HP + 2 * N) {  // 128: rows 2..65, cols 0..1
    const int u = h - 4 * HP;
    y = P + (u >> 1);
    x = u & 1;
  } else {                          // 128: rows 2..65, cols 66..67
    const int u = h - 4 * HP - 2 * N;
    y = P + (u >> 1);
    x = (HP - P) + (u & 1);
  }

  int fA, iA, fB, iB;
  hpx_src(face, y, x, fA, iA, fB, iB);

  const int base = nfc - face * C;  // (b*12)*C + c
  float v = in[(size_t)(base + fA * C) * (size_t)(N * N) + iA];
  if (fB >= 0) {
    v = 0.5f * (v + in[(size_t)(base + fB * C) * (size_t)(N * N) + iB]);
  }
  out[(size_t)nfc * (size_t)(HP * HP) + y * HP + x] = v;
}

// ---------------------------------------------------------------------------
extern "C" void kernel_launch(void* const* d_in, const int* in_sizes, int n_in,
                              void* d_out, int out_size, void* d_ws,
                              size_t ws_size, hipStream_t stream) {
  (void)in_sizes; (void)n_in; (void)out_size; (void)d_ws; (void)ws_size;
  const float* in = (const float*)d_in[0];
  float* out = (float*)d_out;
  // d_in[1] is p (==2), baked into the kernels as constexpr P.

  // Interior: one block per image, all lanes active (async ops need EXEC==~0).
  hpx_interior_kernel<<<NFC, 256, 0, stream>>>(in, out);

  // Halo: 528 cells per image; total divisible by 256 -> full waves only.
  const int halo_total = NFC * HALO;  // 12,976,128
  hpx_halo_kernel<<<halo_total / 256, 256, 0, stream>>>(in, out, halo_total);
}